// CIEDE2000_Loss_23854248362915
// MI455X (gfx1250) — compile-verified
//
#include <hip/hip_runtime.h>

// ---------------------------------------------------------------------------
// CIEDE2000 loss, MI455X (gfx1250, wave32).
// Elementwise transcendental kernel. Roofline: 117 MB streamed (~5us @
// 23.3 TB/s) vs ~50 TRANS ops/pixel -> memory/TRANS balanced. Strategy:
//   * raw v_exp/v_log/v_sin/v_cos/v_sqrt/v_rsq/v_rcp (no OCML call bloat)
//   * custom octant-reduced atan2 polynomial (~15 VALU + 2 v_rcp)
//   * trig multiple-angle identities: the four cosines of T from one
//     v_sin/v_cos pair
//   * B128 nontemporal loads/stores, 4 px/lane, shift/mask indexing
// WMMA intentionally unused: only matrix structure is a 3x3 matvec
// (9 FMAs/px); a 16x16x4 WMMA tile burns 1024 MACs + cross-lane permutes
// for 144 useful MACs -- a strict pessimization for this op.
// ---------------------------------------------------------------------------

typedef float v4f __attribute__((ext_vector_type(4)));

#define DEV static __device__ __forceinline__

// ---- raw gfx1250 hardware math (single v_* instructions) ------------------
DEV float hrcp (float x) { return __builtin_amdgcn_rcpf(x);  }   // v_rcp_f32
DEV float hrsq (float x) { return __builtin_amdgcn_rsqf(x);  }   // v_rsq_f32
DEV float hsqrt(float x) { return __builtin_amdgcn_sqrtf(x); }   // v_sqrt_f32
DEV float hexp2(float x) { return __builtin_amdgcn_exp2f(x); }   // v_exp_f32
DEV float hlog2(float x) { return __builtin_amdgcn_logf(x);  }   // v_log_f32
// v_sin/v_cos consume rotations (x/2pi). Feed degrees/360 directly.
DEV float sind(float deg) { return __builtin_amdgcn_sinf(deg * (1.0f / 360.0f)); }
DEV float cosd(float deg) { return __builtin_amdgcn_cosf(deg * (1.0f / 360.0f)); }
DEV float fdiv(float a, float b) { return a * hrcp(b); }

DEV float pow7f(float x) { float x2 = x * x, x3 = x2 * x; return x3 * x3 * x; }

// sRGB -> linear:  x > 0.04045 ? ((x+0.055)/1.055)^2.4 : x/12.92
// pow via exp2(2.4*log2(x)); argument >= 0.0521 on the used branch.
DEV float srgb_lin(float x) {
    float p = hexp2(2.4f * hlog2((x + 0.055f) * (1.0f / 1.055f)));
    return x > 0.04045f ? p : x * (1.0f / 12.92f);
}

// Lab f():  t > 0.008856 ? cbrt(t) : 7.787 t + 4/29
DEV float lab_f(float t) {
    float c = hexp2((1.0f / 3.0f) * hlog2(t));   // exact-branch arg > 0.008856
    return t > 0.008856f ? c : 7.787f * t + (4.0f / 29.0f);
}

DEV void rgb2lab(float r, float g, float b, float& L, float& A, float& B) {
    float lr = srgb_lin(r), lg = srgb_lin(g), lb = srgb_lin(b);
    float xn = (0.412453f * lr + 0.357580f * lg + 0.180423f * lb) * (1.0f / 0.95047f);
    float yn = (0.212671f * lr + 0.715160f * lg + 0.072169f * lb);
    float zn = (0.019334f * lr + 0.119193f * lg + 0.950227f * lb) * (1.0f / 1.08883f);
    float fx = lab_f(xn), fy = lab_f(yn), fz = lab_f(zn);
    L = 116.0f * fy - 16.0f;
    A = 500.0f * (fx - fy);
    B = 200.0f * (fy - fz);
}

// atan2 in DEGREES, range [-180,180]; octant reduction + degree-9 odd
// minimax (Cephes coeffs, ~1e-7 rad). Matches jnp.arctan2 conventions:
// atan2(0,0)=0, atan2(+-0,-x)=+-180.
DEV float atan2_deg(float y, float x) {
    float ax = __builtin_fabsf(x), ay = __builtin_fabsf(y);
    float mx = __builtin_fmaxf(ax, ay);
    float mn = __builtin_fminf(ax, ay);
    float t  = mx > 0.0f ? mn * hrcp(mx) : 0.0f;        // t in [0,1]
    bool  red = t > 0.4142135624f;                      // tan(pi/8)
    float u  = red ? (t - 1.0f) * hrcp(t + 1.0f) : t;   // fold to [-.414,.414]
    float u2 = u * u;
    float p  = __builtin_fmaf(u2,  8.05374449538e-2f, -1.38776856032e-1f);
    p        = __builtin_fmaf(u2, p,  1.99777106478e-1f);
    p        = __builtin_fmaf(u2, p, -3.33329491539e-1f);
    float r  = __builtin_fmaf(u2 * u, p, u);            // atan(u) rad
    if (red)       r += 0.785398163397f;                // + pi/4
    if (ay > ax)   r  = 1.570796326795f - r;            // reflect over 45deg
    if (x < 0.0f)  r  = 3.141592653590f - r;            // quadrant II
    r = __builtin_copysignf(r, y);                      // quadrants III/IV
    return r * 57.29577951308232f;                      // -> degrees
}

DEV float hue_deg(float b, float ap) {                  // mod 360 -> [0,360)
    float h = atan2_deg(b, ap);
    return h < 0.0f ? h + 360.0f : h;
}

DEV float ciede2000(float L1, float a1, float b1, float L2, float a2, float b2) {
    const float P25_7 = 6103515625.0f;                  // 25^7

    float C1 = hsqrt(a1 * a1 + b1 * b1);
    float C2 = hsqrt(a2 * a2 + b2 * b2);
    float Cbar = 0.5f * (C1 + C2);
    float c7 = pow7f(Cbar);
    float G = 0.5f * (1.0f - hsqrt(fdiv(c7, c7 + P25_7)));

    float a1p = a1 * (1.0f + G);
    float a2p = a2 * (1.0f + G);
    float C1p = hsqrt(a1p * a1p + b1 * b1);
    float C2p = hsqrt(a2p * a2p + b2 * b2);

    float h1 = hue_deg(b1, a1p);
    float h2 = hue_deg(b2, a2p);

    bool zero_chroma = (C1p * C2p) == 0.0f;

    float dLp = L2 - L1;
    float dCp = C2p - C1p;

    float hdiff = h2 - h1;
    float dhp = hdiff;
    if (hdiff > 180.0f)       dhp = hdiff - 360.0f;
    else if (hdiff < -180.0f) dhp = hdiff + 360.0f;
    if (zero_chroma)          dhp = 0.0f;

    float dHp = 2.0f * hsqrt(C1p * C2p) * sind(0.5f * dhp);

    float Lbp = 0.5f * (L1 + L2);
    float Cbp = 0.5f * (C1p + C2p);

    float hsum = h1 + h2;
    float hbp;
    if (__builtin_fabsf(hdiff) > 180.0f)
        hbp = (hsum < 360.0f) ? 0.5f * (hsum + 360.0f) : 0.5f * (hsum - 360.0f);
    else
        hbp = 0.5f * hsum;
    if (zero_chroma) hbp = hsum;

    // T: cos(h-30), cos2h, cos(3h+6), cos(4h-63) from ONE v_sin/v_cos pair.
    float c  = cosd(hbp);
    float s  = sind(hbp);
    float c2 = 2.0f * c * c - 1.0f;   // cos 2h
    float s2 = 2.0f * s * c;          // sin 2h
    float c3 = c * c2 - s * s2;       // cos 3h
    float s3 = s * c2 + c * s2;       // sin 3h
    float c4 = c2 * c2 - s2 * s2;     // cos 4h
    float s4 = 2.0f * s2 * c2;        // sin 4h
    const float cos30 = 0.86602540378443865f, sin30 = 0.5f;
    const float cos6  = 0.99452189536827334f, sin6  = 0.10452846326765347f;
    const float cos63 = 0.45399049973954680f, sin63 = 0.89100652418836786f;
    float T = 1.0f
            - 0.17f * (c * cos30 + s * sin30)    // cos(h - 30)
            + 0.24f * c2                         // cos(2h)
            + 0.32f * (c3 * cos6 - s3 * sin6)    // cos(3h + 6)
            - 0.20f * (c4 * cos63 + s4 * sin63); // cos(4h - 63)

    float u = (hbp - 275.0f) * (1.0f / 25.0f);
    float dtheta = 30.0f * hexp2(-u * u * 1.4426950408889634f); // exp(-u^2)

    float cbp7 = pow7f(Cbp);
    float RC = 2.0f * hsqrt(fdiv(cbp7, cbp7 + P25_7));

    float L50sq = (Lbp - 50.0f) * (Lbp - 50.0f);
    float SL = 1.0f + 0.015f * L50sq * hrsq(20.0f + L50sq);
    float SC = 1.0f + 0.045f * Cbp;
    float SH = 1.0f + 0.015f * Cbp * T;
    float RT = -sind(2.0f * dtheta) * RC;

    float tL = fdiv(dLp, SL);
    float tC = fdiv(dCp, SC);
    float tH = fdiv(dHp, SH);
    return hsqrt(tL * tL + tC * tC + tH * tH + RT * tC * tH);
}

DEV v4f ldnt(const float* p) {
    return __builtin_nontemporal_load((const v4f*)p);  // global_load_b128 th:NT
}
DEV void stnt(float* p, v4f v) {
    __builtin_nontemporal_store(v, (v4f*)p);           // global_store_b128 th:NT
}

// Layout [B,3,H,W], H=W=512: channel stride HW = 2^18, HW/4 = 2^16 ->
// pure shift/mask indexing (no integer divide). One lane = 4 pixels.
__global__ void __launch_bounds__(256)
ciede2000_kernel(const float* __restrict__ img1, const float* __restrict__ img2,
                 float* __restrict__ out, int npix4) {
    constexpr int HW = 512 * 512;          // 2^18
    int i = blockIdx.x * blockDim.x + threadIdx.x;
    if (i >= npix4) return;

    int bi = i >> 16;                      // i / (HW/4)
    int r4 = i & 0xFFFF;                   // i % (HW/4)
    size_t base = (size_t)bi * (3u * HW) + (size_t)r4 * 4u;

    v4f r1  = ldnt(img1 + base);
    v4f g1  = ldnt(img1 + base + HW);
    v4f bb1 = ldnt(img1 + base + 2 * HW);
    v4f r2  = ldnt(img2 + base);
    v4f g2  = ldnt(img2 + base + HW);
    v4f bb2 = ldnt(img2 + base + 2 * HW);

    v4f o;
#pragma unroll
    for (int k = 0; k < 4; ++k) {
        float L1, A1, B1, L2, A2, B2;
        rgb2lab(r1[k], g1[k], bb1[k], L1, A1, B1);
        rgb2lab(r2[k], g2[k], bb2[k], L2, A2, B2);
        o[k] = ciede2000(L1, A1, B1, L2, A2, B2);
    }

    stnt(out + (size_t)i * 4u, o);
}

extern "C" void kernel_launch(void* const* d_in, const int* in_sizes, int n_in,
                              void* d_out, int out_size, void* d_ws, size_t ws_size,
                              hipStream_t stream) {
    const float* img1 = (const float*)d_in[0];
    const float* img2 = (const float*)d_in[1];
    float* out = (float*)d_out;

    int npix4 = out_size >> 2;             // out_size = B*H*W, divisible by 4
    const int threads = 256;               // 8 wave32 waves per block
    int blocks = (npix4 + threads - 1) / threads;

    hipLaunchKernelGGL(ciede2000_kernel, dim3(blocks), dim3(threads), 0, stream,
                       img1, img2, out, npix4);
}